// Sparsify_43353399886084
// MI455X (gfx1250) — compile-verified
//
#include <hip/hip_runtime.h>
#include <stdint.h>

// ---------------------------------------------------------------------------
// Top-K row sparsification (K=256 descending, stable ties) for 4096 x 32768 f32.
// Memory-bound: 1 GB min traffic -> ~43us at 23.3 TB/s. Strategy: stage each
// 128KB row into LDS once via the CDNA5 Tensor Data Mover, do a 3-level
// LDS-resident radix select (12+10+10 bits of an order-preserving key) with
// ds atomics, resolve index-stable ties exactly, then stream the masked row
// back with b128 stores. HBM is touched exactly once per direction.
// ---------------------------------------------------------------------------

#define N_COLS   32768
#define K_TOP    256u
#define BLOCK    1024
#define V4_PT    8          // float4 chunks per thread: 32768 / 4 / 1024

typedef unsigned int v4u __attribute__((ext_vector_type(4)));
typedef int          v8i __attribute__((ext_vector_type(8)));
typedef int          v4i __attribute__((ext_vector_type(4)));

#ifdef __has_builtin
#  if __has_builtin(__builtin_amdgcn_tensor_load_to_lds) && \
      __has_builtin(__builtin_amdgcn_s_wait_tensorcnt)
#    define USE_TDM 1
#  endif
#endif
#ifndef USE_TDM
#  define USE_TDM 0
#endif

// Order-preserving float -> u32 key (larger float => larger key).
__device__ __forceinline__ uint32_t f2key(float f) {
    uint32_t u = __float_as_uint(f);
    return u ^ ((uint32_t)((int32_t)u >> 31) | 0x80000000u);
}

__device__ __forceinline__ float fget(const float4& v, int j) {
    return j == 0 ? v.x : (j == 1 ? v.y : (j == 2 ? v.z : v.w));
}

// Suffix-sum aux[0..1023] in place (Hillis-Steele), then find the unique
// bucket t with C[t] >= Kt > C[t+1]. Writes bucket to *out_b, count of
// elements in strictly-higher buckets to *out_g.
__device__ __forceinline__ void suffix_pick_1024(uint32_t* aux, uint32_t Kt,
                                                 uint32_t* out_b, uint32_t* out_g,
                                                 int tid) {
    for (int off = 1; off < 1024; off <<= 1) {
        uint32_t v = aux[tid];
        if (tid + off < 1024) v += aux[tid + off];
        __syncthreads();
        aux[tid] = v;
        __syncthreads();
    }
    uint32_t c  = aux[tid];
    uint32_t cn = (tid == 1023) ? 0u : aux[tid + 1];
    if (c >= Kt && cn < Kt) { *out_b = (uint32_t)tid; *out_g = cn; }
    __syncthreads();
}

__global__ void __launch_bounds__(BLOCK)
topk_sparsify_kernel(const float* __restrict__ in, float* __restrict__ out) {
    const int row = blockIdx.x;
    const int tid = threadIdx.x;

    extern __shared__ unsigned char smem[];
    float*    rowBuf = (float*)smem;                       // 131072 B
    uint32_t* hist   = (uint32_t*)(smem + 131072);         //  16384 B (4096 buckets)
    uint32_t* aux    = (uint32_t*)(smem + 131072 + 16384); //   4096 B (1024 scan)
    uint32_t* sv     = (uint32_t*)(smem + 131072 + 16384 + 4096); // scalars

    const float4* rowBuf4 = (const float4*)rowBuf;
    const float*  rowIn   = in + (size_t)row * N_COLS;

    // ---------------- Phase 0: stage row into LDS (TDM) ----------------
#if USE_TDM
    if (tid < 32) {
        uint64_t ga     = (uint64_t)(uintptr_t)rowIn;
        uint32_t ldsOff = (uint32_t)(uintptr_t)rowBuf;   // LDS byte offset (low 32 of flat addr)
        v4u g0;
        g0[0] = 1u;                                           // count=1 (valid D#)
        g0[1] = ldsOff;                                       // lds_addr [63:32]
        g0[2] = (uint32_t)ga;                                 // global_addr [95:64]
        g0[3] = (uint32_t)((ga >> 32) & 0x1FFFFFFu) | (2u << 30); // addr[56:32] | type=2
        v8i g1;
        g1[0] = (int)(2u << 16);                              // data_size = 4B
        g1[1] = (int)(((uint32_t)N_COLS & 0xFFFFu) << 16);    // tensor_dim0 lo16
        g1[2] = (int)((((uint32_t)N_COLS >> 16) & 0xFFFFu) | (1u << 16)); // dim0 hi | tensor_dim1=1 lo
        g1[3] = (int)((uint32_t)N_COLS << 16);                // tile_dim0 = 32768
        g1[4] = 1;                                            // tile_dim1 = 1, tile_dim2 = 0
        g1[5] = (int)(uint32_t)N_COLS;                        // tensor_dim0_stride lo32
        g1[6] = 0;
        g1[7] = 0;
        v4i gz4 = {0, 0, 0, 0};
        v8i gz8 = {0, 0, 0, 0, 0, 0, 0, 0};
        // clang-23 / therock form: 6 args (g0, g1, g2, g3, g4, cpol)
        __builtin_amdgcn_tensor_load_to_lds(g0, g1, gz4, gz4, gz8, 0);
        __builtin_amdgcn_s_wait_tensorcnt(0);
    }
#else
    for (int k = 0; k < V4_PT; ++k) {
        int i4 = tid + k * BLOCK;
        ((float4*)rowBuf)[i4] = ((const float4*)rowIn)[i4];
    }
#endif
    __syncthreads();

    // ---------------- Phase 1: 12-bit histogram (key[31:20]) ----------------
    for (int j = tid; j < 4096; j += BLOCK) hist[j] = 0u;
    __syncthreads();
    for (int k = 0; k < V4_PT; ++k) {
        float4 v = rowBuf4[tid + k * BLOCK];
        atomicAdd(&hist[f2key(v.x) >> 20], 1u);
        atomicAdd(&hist[f2key(v.y) >> 20], 1u);
        atomicAdd(&hist[f2key(v.z) >> 20], 1u);
        atomicAdd(&hist[f2key(v.w) >> 20], 1u);
    }
    __syncthreads();
    aux[tid] = hist[4 * tid] + hist[4 * tid + 1] + hist[4 * tid + 2] + hist[4 * tid + 3];
    __syncthreads();
    suffix_pick_1024(aux, K_TOP, &sv[0], &sv[1], tid);
    if (tid == 0) {   // refine within the group of 4 buckets
        uint32_t g = sv[0], cum = sv[1], b1s = 0, G1s = 0;
        for (int b = (int)g * 4 + 3; b >= (int)g * 4; --b) {
            uint32_t h = hist[b];
            if (cum + h >= K_TOP) { b1s = (uint32_t)b; G1s = cum; break; }
            cum += h;
        }
        sv[2] = b1s; sv[3] = G1s;
    }
    __syncthreads();
    const uint32_t b1 = sv[2], G1 = sv[3];

    // ---------------- Phase 2: 10-bit refine (key[19:10]) ----------------
    for (int j = tid; j < 1024; j += BLOCK) hist[j] = 0u;
    __syncthreads();
    for (int k = 0; k < V4_PT; ++k) {
        float4 v = rowBuf4[tid + k * BLOCK];
        #pragma unroll
        for (int j = 0; j < 4; ++j) {
            uint32_t key = f2key(fget(v, j));
            if ((key >> 20) == b1) atomicAdd(&hist[(key >> 10) & 1023u], 1u);
        }
    }
    __syncthreads();
    aux[tid] = hist[tid];
    __syncthreads();
    const uint32_t K2 = K_TOP - G1;
    suffix_pick_1024(aux, K2, &sv[0], &sv[1], tid);
    const uint32_t b2 = sv[0], G2 = sv[1];

    // ---------------- Phase 3: 10-bit refine (key[9:0]) ----------------
    const uint32_t pre22 = (b1 << 10) | b2;
    for (int j = tid; j < 1024; j += BLOCK) hist[j] = 0u;
    __syncthreads();
    for (int k = 0; k < V4_PT; ++k) {
        float4 v = rowBuf4[tid + k * BLOCK];
        #pragma unroll
        for (int j = 0; j < 4; ++j) {
            uint32_t key = f2key(fget(v, j));
            if ((key >> 10) == pre22) atomicAdd(&hist[key & 1023u], 1u);
        }
    }
    __syncthreads();
    aux[tid] = hist[tid];
    __syncthreads();
    const uint32_t K3 = K2 - G2;
    suffix_pick_1024(aux, K3, &sv[0], &sv[1], tid);
    if (tid == 0) {
        uint32_t b3 = sv[0], G3 = sv[1];
        sv[2] = (pre22 << 10) | b3;   // exact threshold key T
        sv[3] = K3 - G3;              // ties to keep
        sv[4] = hist[b3];             // total ties at T
    }
    __syncthreads();
    const uint32_t T = sv[2], tiesKeep = sv[3], tieAll = sv[4];

    // ------------- Tie resolution: stable (lowest-index ties kept) -------------
    uint32_t idxCut;
    if (tiesKeep >= tieAll) {
        idxCut = 0xFFFFFFFFu;   // keep all ties
    } else {
        // MSB-first radix select of the tiesKeep-th smallest index among key==T.
        if (tid == 0) { sv[5] = 0u; sv[6] = tiesKeep; }
        __syncthreads();
        for (int bit = 14; bit >= 0; --bit) {
            if (tid == 0) sv[7] = 0u;
            __syncthreads();
            uint32_t p = sv[5];
            uint32_t local = 0;
            for (int k = 0; k < V4_PT; ++k) {
                int i4 = tid + k * BLOCK;
                float4 v = rowBuf4[i4];
                uint32_t base = (uint32_t)i4 * 4u;
                #pragma unroll
                for (int j = 0; j < 4; ++j) {
                    uint32_t key = f2key(fget(v, j));
                    uint32_t idx = base + (uint32_t)j;
                    if (key == T && (idx >> (bit + 1)) == p && ((idx >> bit) & 1u) == 0u)
                        ++local;
                }
            }
            if (local) atomicAdd(&sv[7], local);
            __syncthreads();
            if (tid == 0) {
                uint32_t c0 = sv[7], R = sv[6];
                if (R <= c0) sv[5] = p << 1;
                else { sv[6] = R - c0; sv[5] = (p << 1) | 1u; }
            }
            __syncthreads();
        }
        idxCut = sv[5];
    }

    // ---------------- Output: masked row, coalesced b128 stores ----------------
    float4* out4 = (float4*)(out + (size_t)row * N_COLS);
    for (int k = 0; k < V4_PT; ++k) {
        int i4 = tid + k * BLOCK;
        float4 v = rowBuf4[i4];
        uint32_t base = (uint32_t)i4 * 4u;
        float4 r;
        {
            uint32_t key = f2key(v.x);
            r.x = (key > T || (key == T && base + 0u <= idxCut)) ? v.x : 0.0f;
        }
        {
            uint32_t key = f2key(v.y);
            r.y = (key > T || (key == T && base + 1u <= idxCut)) ? v.y : 0.0f;
        }
        {
            uint32_t key = f2key(v.z);
            r.z = (key > T || (key == T && base + 2u <= idxCut)) ? v.z : 0.0f;
        }
        {
            uint32_t key = f2key(v.w);
            r.w = (key > T || (key == T && base + 3u <= idxCut)) ? v.w : 0.0f;
        }
        out4[i4] = r;
    }
}

extern "C" void kernel_launch(void* const* d_in, const int* in_sizes, int n_in,
                              void* d_out, int out_size, void* d_ws, size_t ws_size,
                              hipStream_t stream) {
    (void)n_in; (void)out_size; (void)d_ws; (void)ws_size;
    const float* in  = (const float*)d_in[0];
    float*       out = (float*)d_out;
    const int rows = in_sizes[0] / N_COLS;   // 4096

    // 128KB row + 16KB hist + 4KB scan + 64B scalars of dynamic LDS
    const size_t shmem = 131072 + 16384 + 4096 + 64;
    // Opt in to >64KB dynamic LDS (gfx1250 WGP has 320KB). Deterministic,
    // not a stream op, safe under graph capture.
    (void)hipFuncSetAttribute((const void*)topk_sparsify_kernel,
                              hipFuncAttributeMaxDynamicSharedMemorySize, (int)shmem);

    topk_sparsify_kernel<<<dim3(rows), dim3(BLOCK), shmem, stream>>>(in, out);
}